// Permute_87926570483907
// MI455X (gfx1250) — compile-verified
//
#include <hip/hip_runtime.h>

// CDNA5 / gfx1250: wave32, WMMA f32 16x16x4.
typedef __attribute__((ext_vector_type(2))) float v2f;
typedef __attribute__((ext_vector_type(8))) float v8f;

#define DIM    192
#define HW     16384            // 128*128 spatial positions per (batch, channel)
#define GTILES 8                // spatial 16-wide n-tiles per wave

// Block: 384 threads = 12 waves. Wave w computes output channels [16w, 16w+16)
// for 128 consecutive spatial columns (8 WMMA n-tiles). One block therefore
// produces all 192 output channels for its spatial slice -> x is read from HBM
// exactly once; the 12-way cross-wave B reuse hits WGP$/L2. P rows are tiny
// (12 KB per wave) and L0-resident, loaded directly from global each K-step.
__global__ __launch_bounds__(384) void permute_wmma_kernel(
    const float* __restrict__ x,   // (16, 192, 128, 128)
    const float* __restrict__ P,   // (192, 192) row-major
    float* __restrict__ y)         // (16, 192, 128, 128)
{
    const unsigned tid   = threadIdx.x;
    const unsigned lane  = tid & 31u;
    const unsigned wave  = tid >> 5;        // 0..11 -> output-channel tile
    const unsigned o0    = wave * 16u;
    const unsigned n     = lane & 15u;      // A-matrix row (M) / B,C,D column (N)
    const unsigned half  = lane >> 4;       // 0/1 -> K sub-offset per ISA layout
    const unsigned s0    = blockIdx.x * (16u * GTILES);
    const unsigned batch = blockIdx.y;

    const float* xb = x + (size_t)batch * (DIM * HW);
    float*       yb = y + (size_t)batch * (DIM * HW);

    v8f acc[GTILES] = {};                   // zero accumulators

    const unsigned scol = s0 + n;           // base spatial column for this lane

    for (unsigned k = 0; k < DIM; k += 4u) {
        // A fragment: 16x4 fp32 tile of P.
        // lane L(0-15):  {P[o0+L, k+0], P[o0+L, k+1]}
        // lane L(16-31): {P[o0+L-16, k+2], P[o0+L-16, k+3]}   (8B aligned)
        v2f a = *(const v2f*)(P + (size_t)(o0 + n) * DIM + (k + 2u * half));

        // B fragment base: X[k + 2*half, scol]; second component is next K row.
        const float* xk = xb + (size_t)(k + 2u * half) * HW + scol;

#pragma unroll
        for (int g = 0; g < GTILES; ++g) {
            v2f b;
            b.x = xk[g * 16];               // K = k + 2*half,     N = scol + 16g
            b.y = xk[g * 16 + HW];          // K = k + 2*half + 1, N = scol + 16g
            // 8-arg form: (neg_a, A, neg_b, B, c_mod, C, reuse_a, reuse_b)
            acc[g] = __builtin_amdgcn_wmma_f32_16x16x4_f32(
                false, a, false, b, (short)0, acc[g], false, false);
        }
    }

    // D layout (32-bit 16x16): lane 0-15 -> N, VGPR r -> M=r (lanes 0-15),
    // M=r+8 (lanes 16-31). Stores: 16 contiguous lanes -> 64B segments.
    // Non-temporal: y is write-once, keep L2 for x.
#pragma unroll
    for (int g = 0; g < GTILES; ++g) {
        const unsigned col = s0 + g * 16 + n;
#pragma unroll
        for (int r = 0; r < 8; ++r) {
            __builtin_nontemporal_store(
                acc[g][r],
                yb + (size_t)(o0 + (unsigned)r + half * 8u) * HW + col);
        }
    }
}

__global__ void copy_logv_kernel(const float* __restrict__ logv,
                                 float* __restrict__ out)
{
    if (threadIdx.x < 16) out[threadIdx.x] = logv[threadIdx.x];
}

extern "C" void kernel_launch(void* const* d_in, const int* in_sizes, int n_in,
                              void* d_out, int out_size, void* d_ws, size_t ws_size,
                              hipStream_t stream) {
    const float* x    = (const float*)d_in[0];  // (16,192,128,128) fp32
    const float* logv = (const float*)d_in[1];  // (16,) fp32
    const float* P    = (const float*)d_in[2];  // (192,192) fp32
    float* y = (float*)d_out;

    // 16384 spatial / (16 cols * 8 tiles) = 128 blocks per batch; 16 batches.
    dim3 grid(128, 16);
    permute_wmma_kernel<<<grid, 384, 0, stream>>>(x, P, y);

    // logv pass-through appended after y.
    copy_logv_kernel<<<1, 16, 0, stream>>>(logv, y + (size_t)16 * DIM * HW);
}